// Coulomb_37022618091781
// MI455X (gfx1250) — compile-verified
//
#include <hip/hip_runtime.h>

#define TPB  256
#define TILE 2048   // pairs per tile; 2 int32 each -> 16 KB per LDS buffer
#define NBLK 1024

typedef int v4i __attribute__((vector_size(16)));
typedef __attribute__((address_space(1))) v4i* gv4i_p;
typedef __attribute__((address_space(3))) v4i* lv4i_p;

// ---- CDNA5 async global->LDS copy of one 16B chunk (2 pairs) --------------
__device__ __forceinline__ void async_copy16(const int* gsrc, int* ldst) {
#if defined(__gfx1250__) && __has_builtin(__builtin_amdgcn_global_load_async_to_lds_b128)
  __builtin_amdgcn_global_load_async_to_lds_b128(
      (gv4i_p)gsrc, (lv4i_p)ldst, 0, 0);
#else
  struct alignas(16) P { int a, b, c, d; };
  *(P*)ldst = *(const P*)gsrc;
#endif
}

__device__ __forceinline__ void wait_async_lds() {
#if defined(__gfx1250__)
#if __has_builtin(__builtin_amdgcn_s_wait_asynccnt)
  __builtin_amdgcn_s_wait_asynccnt(0);
#else
  asm volatile("s_wait_asynccnt 0" ::: "memory");
#endif
#endif
}

// ---- main pair kernel: per-block partial sums -----------------------------
__global__ __launch_bounds__(TPB) void Coulomb_pairs_kernel(
    const float* __restrict__ coords,
    const int*   __restrict__ pairs,     // (nPairs, 2) int32
    const float* __restrict__ box,       // 3x3 row-major
    const float* __restrict__ charges,
    const float* __restrict__ cutoffp,
    const int*   __restrict__ do_shiftp,
    float*       __restrict__ partial,
    long long nPairs)
{
  __shared__ alignas(16) int sp[2][2 * TILE];
  __shared__ float sred[TPB];

  // uniform 3x3 box inverse
  const float b00 = box[0], b01 = box[1], b02 = box[2];
  const float b10 = box[3], b11 = box[4], b12 = box[5];
  const float b20 = box[6], b21 = box[7], b22 = box[8];
  const float det = b00*(b11*b22 - b12*b21) - b01*(b10*b22 - b12*b20) + b02*(b10*b21 - b11*b20);
  const float dv  = 1.0f / det;
  const float i00 = (b11*b22 - b12*b21)*dv, i01 = (b02*b21 - b01*b22)*dv, i02 = (b01*b12 - b02*b11)*dv;
  const float i10 = (b12*b20 - b10*b22)*dv, i11 = (b00*b22 - b02*b20)*dv, i12 = (b02*b10 - b00*b12)*dv;
  const float i20 = (b10*b21 - b11*b20)*dv, i21 = (b01*b20 - b00*b21)*dv, i22 = (b00*b11 - b01*b10)*dv;

  const float cutoff  = cutoffp[0];
  const float cutoff2 = cutoff * cutoff;
  const float shift   = (do_shiftp[0] != 0) ? (1.0f / cutoff) : 0.0f;

  const long long nTiles = (nPairs + TILE - 1) / TILE;
  float acc = 0.0f;

  const long long tile0 = blockIdx.x;
  if (tile0 < nTiles) {
    // stage first tile into buffer 0 (async)
    {
      const long long base2 = tile0 * (2LL * TILE);   // int32 offset of tile
      #pragma unroll
      for (int u = 0; u < TILE / 2 / TPB; ++u) {      // 16B chunks = 2 pairs each
        const int c = u * TPB + threadIdx.x;          // chunk index within tile
        const long long gp = tile0 * TILE + 2LL * c;  // first pair of chunk
        if (gp < nPairs) async_copy16(pairs + base2 + 4LL * c, &sp[0][4 * c]);
      }
    }
    int cur = 0;
    for (long long tile = tile0; tile < nTiles; tile += gridDim.x) {
      wait_async_lds();     // drain this wave's async loads
      __syncthreads();      // whole tile staged + prior readers done

      // stage next tile into the other buffer while we compute
      const long long nxt = tile + gridDim.x;
      if (nxt < nTiles) {
        const long long base2 = nxt * (2LL * TILE);
        #pragma unroll
        for (int u = 0; u < TILE / 2 / TPB; ++u) {
          const int c = u * TPB + threadIdx.x;
          const long long gp = nxt * TILE + 2LL * c;
          if (gp < nPairs) async_copy16(pairs + base2 + 4LL * c, &sp[cur ^ 1][4 * c]);
        }
      }
      // speculative prefetch two tiles ahead (gfx1250 global_prefetch_b8)
      const long long nxt2 = tile + 2LL * gridDim.x;
      if (nxt2 < nTiles) {
        __builtin_prefetch((const char*)(pairs + nxt2 * (2LL * TILE)) +
                           (size_t)threadIdx.x * 128, 0, 0);
      }

      // compute on current buffer (L2-resident gathers)
      const long long base = tile * TILE;
      #pragma unroll
      for (int u = 0; u < TILE / TPB; ++u) {
        const int t = u * TPB + threadIdx.x;
        const long long idx = base + t;
        if (idx < nPairs) {
          const int i = sp[cur][2 * t];
          const int j = sp[cur][2 * t + 1];
          const float cix = coords[3 * i], ciy = coords[3 * i + 1], ciz = coords[3 * i + 2];
          const float cjx = coords[3 * j], cjy = coords[3 * j + 1], cjz = coords[3 * j + 2];
          const float dx = cix - cjx, dy = ciy - cjy, dz = ciz - cjz;
          float sx = dx * i00 + dy * i10 + dz * i20;
          float sy = dx * i01 + dy * i11 + dz * i21;
          float sz = dx * i02 + dy * i12 + dz * i22;
          sx -= floorf(sx + 0.5f);
          sy -= floorf(sy + 0.5f);
          sz -= floorf(sz + 0.5f);
          const float px = sx * b00 + sy * b10 + sz * b20;
          const float py = sx * b01 + sy * b11 + sz * b21;
          const float pz = sx * b02 + sy * b12 + sz * b22;
          const float r2 = px * px + py * py + pz * pz;
          if (r2 <= cutoff2) {
            const float rinv = 1.0f / sqrtf(r2);
            acc += charges[i] * charges[j] * (rinv - shift);
          }
        }
      }
      cur ^= 1;
    }
  }

  // deterministic block reduction
  sred[threadIdx.x] = acc;
  __syncthreads();
  #pragma unroll
  for (int s = TPB / 2; s > 0; s >>= 1) {
    if ((int)threadIdx.x < s) sred[threadIdx.x] += sred[threadIdx.x + s];
    __syncthreads();
  }
  if (threadIdx.x == 0) partial[blockIdx.x] = sred[0];
}

// ---- deterministic final reduction ----------------------------------------
__global__ __launch_bounds__(TPB) void Coulomb_reduce_kernel(
    const float* __restrict__ partial, int n,
    const float* __restrict__ prefacp, float* __restrict__ out)
{
  __shared__ float s[TPB];
  float a = 0.0f;
  for (int k = threadIdx.x; k < n; k += TPB) a += partial[k];  // fixed order
  s[threadIdx.x] = a;
  __syncthreads();
  #pragma unroll
  for (int st = TPB / 2; st > 0; st >>= 1) {
    if ((int)threadIdx.x < st) s[threadIdx.x] += s[threadIdx.x + st];
    __syncthreads();
  }
  if (threadIdx.x == 0) out[0] = s[0] * prefacp[0];
}

extern "C" void kernel_launch(void* const* d_in, const int* in_sizes, int n_in,
                              void* d_out, int out_size, void* d_ws, size_t ws_size,
                              hipStream_t stream) {
  (void)n_in; (void)out_size; (void)ws_size;
  const float* coords   = (const float*)d_in[0];
  const int*   pairs    = (const int*)d_in[1];   // int32 (JAX x64-off), 2 per pair
  const float* box      = (const float*)d_in[2];
  const float* charges  = (const float*)d_in[3];
  const float* prefac   = (const float*)d_in[4];
  const float* cutoff   = (const float*)d_in[5];
  const int*   do_shift = (const int*)d_in[6];
  const long long nPairs = (long long)in_sizes[1] / 2;

  float* partial = (float*)d_ws;  // NBLK floats

  Coulomb_pairs_kernel<<<NBLK, TPB, 0, stream>>>(coords, pairs, box, charges,
                                                 cutoff, do_shift, partial, nPairs);
  Coulomb_reduce_kernel<<<1, TPB, 0, stream>>>(partial, NBLK, prefac, (float*)d_out);
}